// GAT_62045097558453
// MI455X (gfx1250) — compile-verified
//
#include <hip/hip_runtime.h>
#include <hip/hip_bf16.h>
#include <math.h>

// ---------------------------------------------------------------------------
// CDNA5 WMMA f32 16x16x4 fragments: A 16x4 (2 f32/lane), B 4x16 (2 f32/lane),
// C/D 16x16 (8 f32/lane).  Wave32 only.
// ---------------------------------------------------------------------------
typedef __attribute__((ext_vector_type(2))) float v2f;
typedef __attribute__((ext_vector_type(8))) float v8f;

// flattened itertools.combinations(range(5),2)
constexpr int PI[20] = {0,1,0,2,0,3,0,4,1,2,1,3,1,4,2,3,2,4,3,4};

// ---------------------------------------------------------------------------
// Stage A: per-chunk tiny MHA (heads=5, key_dim=2) over U=192, fused with
// mean-over-edges, GroupNorm (per (b,c) over U, eps=1e-3) and pos-emb.
// One block per (b,c); 192 threads (u index).
// ---------------------------------------------------------------------------
__global__ __launch_bounds__(192)
void chunk_mha_gnorm(const float* __restrict__ x,
                     const float* __restrict__ Wq, const float* __restrict__ bq,
                     const float* __restrict__ Wk, const float* __restrict__ bk,
                     const float* __restrict__ Wv, const float* __restrict__ bv,
                     const float* __restrict__ Wo, const float* __restrict__ bo,
                     const float* __restrict__ pos,
                     const float* __restrict__ gng, const float* __restrict__ gnb,
                     float* __restrict__ Xg) {
  const int bc = blockIdx.x;           // 0..799
  const int b  = bc / 200;
  const int c  = bc % 200;
  const int u  = threadIdx.x;          // 0..191

  __shared__ float Wq_s[200], Wk_s[200], Wv_s[200], Wo_s[200];
  __shared__ float bq_s[10], bk_s[10], bv_s[10], bo_s[20];
  __shared__ float k_s[192 * 10], v_s[192 * 10];
  __shared__ float red1[192], red2[192];
  __shared__ float mu_s, var_s;

  for (int idx = u; idx < 200; idx += 192) {
    Wq_s[idx] = Wq[c * 200 + idx];
    Wk_s[idx] = Wk[c * 200 + idx];
    Wv_s[idx] = Wv[c * 200 + idx];
    Wo_s[idx] = Wo[c * 200 + idx];
  }
  if (u < 10) { bq_s[u] = bq[c * 10 + u]; bk_s[u] = bk[c * 10 + u]; bv_s[u] = bv[c * 10 + u]; }
  if (u < 20) { bo_s[u] = bo[c * 20 + u]; }

  // ch[g] = x[b, c*5+g, u]
  float ch[5];
#pragma unroll
  for (int g = 0; g < 5; ++g)
    ch[g] = x[((size_t)b * 1000 + (size_t)c * 5 + g) * 192 + u];

  __syncthreads();

  // q,k,v projections over the 20 edge channels (h[...,d] = ch[PI[d]])
  float q[10];
#pragma unroll
  for (int hk = 0; hk < 10; ++hk) {
    float aq = bq_s[hk], ak = bk_s[hk], av = bv_s[hk];
#pragma unroll
    for (int d = 0; d < 20; ++d) {
      const float hv = ch[PI[d]];
      aq += hv * Wq_s[d * 10 + hk];
      ak += hv * Wk_s[d * 10 + hk];
      av += hv * Wv_s[d * 10 + hk];
    }
    q[hk] = aq;
    k_s[u * 10 + hk] = ak;
    v_s[u * 10 + hk] = av;
  }
  __syncthreads();

  // attention: per head softmax over 192 source rows (scale 1/sqrt(2))
  const float isq2 = 0.70710678118654752f;
  float ctx[10];
#pragma unroll
  for (int hd = 0; hd < 5; ++hd) {
    const float q0 = q[2 * hd] * isq2, q1 = q[2 * hd + 1] * isq2;
    float mx = -1e30f;
    for (int vv = 0; vv < 192; ++vv)
      mx = fmaxf(mx, q0 * k_s[vv * 10 + 2 * hd] + q1 * k_s[vv * 10 + 2 * hd + 1]);
    float sm = 0.f, c0 = 0.f, c1 = 0.f;
    for (int vv = 0; vv < 192; ++vv) {
      const float e = expf(q0 * k_s[vv * 10 + 2 * hd] + q1 * k_s[vv * 10 + 2 * hd + 1] - mx);
      sm += e;
      c0 += e * v_s[vv * 10 + 2 * hd];
      c1 += e * v_s[vv * 10 + 2 * hd + 1];
    }
    ctx[2 * hd]     = c0 / sm;
    ctx[2 * hd + 1] = c1 / sm;
  }

  // alpha = h + ctx@Wo + bo ; xg = mean over 20 channels
  float xs = 0.f;
#pragma unroll
  for (int d = 0; d < 20; ++d) {
    float a = ch[PI[d]] + bo_s[d];
#pragma unroll
    for (int hk = 0; hk < 10; ++hk) a += ctx[hk] * Wo_s[hk * 20 + d];
    xs += a;
  }
  xs *= 0.05f;

  // GroupNorm over u (biased var), then + pos_emb
  red1[u] = xs; red2[u] = xs * xs;
  __syncthreads();
  for (int st = 96; st >= 3; st >>= 1) {
    if (u < st) { red1[u] += red1[u + st]; red2[u] += red2[u + st]; }
    __syncthreads();
  }
  if (u == 0) {
    const float su = red1[0] + red1[1] + red1[2];
    const float sq = red2[0] + red2[1] + red2[2];
    mu_s  = su / 192.f;
    var_s = sq / 192.f - mu_s * mu_s;
  }
  __syncthreads();

  const float y = (xs - mu_s) * rsqrtf(var_s + 1e-3f) * gng[c] + gnb[c] + pos[c * 192 + u];
  Xg[(size_t)bc * 192 + u] = y;
}

// ---------------------------------------------------------------------------
// f32 WMMA GEMM:  Y[M,N] = act(X[M,K] @ W[K,N] + bias[N])
// Requires M % 32 == 0, N % 64 == 0, K % 16 == 0 (true for all uses here:
// M=800, N=192, K=192).  Guard-free straight-line loads so the compiler can
// batch VMEM and issue 16 back-to-back WMMAs per K-chunk.
// Block = 64 threads (2 waves); wave w computes 16x64 tile at
// rows (blockIdx.y*2 + w)*16, cols blockIdx.x*64.
// ACT: 0 = identity, 1 = exact GELU.
// ---------------------------------------------------------------------------
template <int ACT>
__global__ __launch_bounds__(64)
void wmma_gemm_bias(const float* __restrict__ X, const float* __restrict__ W,
                    const float* __restrict__ bias, float* __restrict__ Y,
                    int M, int N, int K) {
  const int wave = threadIdx.x >> 5;
  const int lane = threadIdx.x & 31;
  const int half = lane >> 4;
  const int ln   = lane & 15;
  const int tm   = blockIdx.y * 2 + wave;  // 16-row tile index
  const int n0   = blockIdx.x * 64;        // 64-col panel start
  const int m    = tm * 16 + ln;           // A row this lane streams
  const int koff = half * 2;

  const float* __restrict__ Arow = X + (size_t)m * K;

  v8f acc[4];
#pragma unroll
  for (int t = 0; t < 4; ++t) {
    const float bval = bias[n0 + t * 16 + ln];
#pragma unroll
    for (int j = 0; j < 8; ++j) acc[t][j] = bval;
  }

  for (int k0 = 0; k0 < K; k0 += 16) {
    v2f a[4];
    v2f bfr[4][4];
#pragma unroll
    for (int kk = 0; kk < 4; ++kk) {
      const int k = k0 + kk * 4 + koff;
      a[kk].x = Arow[k];
      a[kk].y = Arow[k + 1];
#pragma unroll
      for (int t = 0; t < 4; ++t) {
        bfr[kk][t].x = W[(size_t)k * N + n0 + t * 16 + ln];
        bfr[kk][t].y = W[(size_t)(k + 1) * N + n0 + t * 16 + ln];
      }
    }
#pragma unroll
    for (int kk = 0; kk < 4; ++kk)
#pragma unroll
      for (int t = 0; t < 4; ++t)
        acc[t] = __builtin_amdgcn_wmma_f32_16x16x4_f32(
            false, a[kk], false, bfr[kk][t], (short)0, acc[t], false, false);
  }

#pragma unroll
  for (int t = 0; t < 4; ++t)
#pragma unroll
    for (int j = 0; j < 8; ++j) {
      const int row = tm * 16 + j + half * 8;
      float v = acc[t][j];
      if (ACT == 1) v = 0.5f * v * (1.f + erff(v * 0.70710678118654752f));
      Y[(size_t)row * N + n0 + t * 16 + ln] = v;
    }
}

// ---------------------------------------------------------------------------
// Graph attention core: per (row-tile i, head h, batch b) compute
// P = softmax(Qh @ Kh^T / sqrt(48)), then Ctx = P @ Vh via WMMA.
// T=200 tokens padded to 208.  Out-of-range token indices are CLAMPED to 199
// (always-in-bounds straight-line loads, no EXEC games); padded score columns
// are masked to -1e30 before softmax (exp -> exactly 0), and padded rows are
// never stored, so clamping cannot affect the valid outputs.
// One wave (32 threads) per block.
// ---------------------------------------------------------------------------
__global__ __launch_bounds__(32)
void graph_attn(const float* __restrict__ Q, const float* __restrict__ Km,
                const float* __restrict__ V, float* __restrict__ Ctx) {
  const int i = blockIdx.x;   // row tile 0..12
  const int h = blockIdx.y;   // head 0..3
  const int b = blockIdx.z;   // batch 0..3
  const int lane = threadIdx.x;
  const int half = lane >> 4;
  const int ln   = lane & 15;
  const int koff = half * 2;
  const int U = 192, T = 200;
  const float scale = 0.14433756729740643f;  // 1/sqrt(48)

  __shared__ float P[16][208];

  v8f s[13];
  const int qr = min(i * 16 + ln, T - 1);  // clamped query row
  const float* __restrict__ Qrow = Q + ((size_t)(b * 200 + qr)) * U + h * 48;

  // ---- scores: 13 column tiles, K=48 ----
#pragma unroll
  for (int j = 0; j < 13; ++j) {
    v8f c;
#pragma unroll
    for (int r = 0; r < 8; ++r) c[r] = 0.f;
    const int kr = min(j * 16 + ln, T - 1);  // clamped key row
    const float* __restrict__ Krow = Km + ((size_t)(b * 200 + kr)) * U + h * 48;
#pragma unroll
    for (int k0 = 0; k0 < 48; k0 += 4) {
      v2f a, bb;
      a.x = Qrow[k0 + koff];
      a.y = Qrow[k0 + koff + 1];
      bb.x = Krow[k0 + koff];
      bb.y = Krow[k0 + koff + 1];
      c = __builtin_amdgcn_wmma_f32_16x16x4_f32(false, a, false, bb, (short)0, c,
                                                false, false);
    }
    const int col = j * 16 + ln;
#pragma unroll
    for (int r = 0; r < 8; ++r) s[j][r] = (col < T) ? c[r] * scale : -1e30f;
  }

  // ---- softmax per row (cross-lane over the 16 lanes of each half) ----
#pragma unroll
  for (int r = 0; r < 8; ++r) {
    float mx = -1e30f;
#pragma unroll
    for (int j = 0; j < 13; ++j) mx = fmaxf(mx, s[j][r]);
    for (int msk = 1; msk < 16; msk <<= 1) mx = fmaxf(mx, __shfl_xor(mx, msk, 32));
    float sm = 0.f;
#pragma unroll
    for (int j = 0; j < 13; ++j) {
      const float e = expf(s[j][r] - mx);
      s[j][r] = e;
      sm += e;
    }
    for (int msk = 1; msk < 16; msk <<= 1) sm += __shfl_xor(sm, msk, 32);
    const float inv = 1.f / sm;
#pragma unroll
    for (int j = 0; j < 13; ++j) s[j][r] *= inv;
  }

  // ---- stage P through LDS to reach the WMMA A-layout ----
#pragma unroll
  for (int j = 0; j < 13; ++j)
#pragma unroll
    for (int r = 0; r < 8; ++r)
      P[r + 8 * half][j * 16 + ln] = s[j][r];
  __syncthreads();

  // ---- Ctx = P[16,208] @ Vh[208,48] ----
#pragma unroll
  for (int nt = 0; nt < 3; ++nt) {
    v8f c;
#pragma unroll
    for (int r = 0; r < 8; ++r) c[r] = 0.f;
    for (int k0 = 0; k0 < 208; k0 += 4) {
      v2f a, bb;
      a.x = P[ln][k0 + koff];
      a.y = P[ln][k0 + koff + 1];
      const int vr0 = min(k0 + koff, T - 1);      // clamped value rows
      const int vr1 = min(k0 + koff + 1, T - 1);  // (P is exactly 0 there)
      bb.x = V[((size_t)(b * 200 + vr0)) * U + h * 48 + nt * 16 + ln];
      bb.y = V[((size_t)(b * 200 + vr1)) * U + h * 48 + nt * 16 + ln];
      c = __builtin_amdgcn_wmma_f32_16x16x4_f32(false, a, false, bb, (short)0, c,
                                                false, false);
    }
#pragma unroll
    for (int r = 0; r < 8; ++r) {
      const int t = i * 16 + r + 8 * half;
      if (t < T)
        Ctx[((size_t)(b * 200 + t)) * U + h * 48 + nt * 16 + ln] = c[r];
    }
  }
}

// ---------------------------------------------------------------------------
// Residual + LayerNorm over last dim (192), eps=1e-3.  One row per block.
// ---------------------------------------------------------------------------
__global__ __launch_bounds__(192)
void resid_layernorm(const float* __restrict__ A, const float* __restrict__ Bm,
                     const float* __restrict__ g, const float* __restrict__ be,
                     float* __restrict__ Y) {
  const int r = blockIdx.x;
  const int u = threadIdx.x;
  __shared__ float s1[192], s2[192];
  __shared__ float mu_s, var_s;

  const float xv = A[(size_t)r * 192 + u] + Bm[(size_t)r * 192 + u];
  s1[u] = xv; s2[u] = xv * xv;
  __syncthreads();
  for (int st = 96; st >= 3; st >>= 1) {
    if (u < st) { s1[u] += s1[u + st]; s2[u] += s2[u + st]; }
    __syncthreads();
  }
  if (u == 0) {
    const float su = s1[0] + s1[1] + s1[2];
    const float sq = s2[0] + s2[1] + s2[2];
    mu_s  = su / 192.f;
    var_s = sq / 192.f - mu_s * mu_s;
  }
  __syncthreads();
  Y[(size_t)r * 192 + u] = (xv - mu_s) * rsqrtf(var_s + 1e-3f) * g[u] + be[u];
}

// ---------------------------------------------------------------------------
// Launch
// ---------------------------------------------------------------------------
extern "C" void kernel_launch(void* const* d_in, const int* in_sizes, int n_in,
                              void* d_out, int out_size, void* d_ws, size_t ws_size,
                              hipStream_t stream) {
  const float* x    = (const float*)d_in[0];
  const float* Wq   = (const float*)d_in[1];
  const float* bq   = (const float*)d_in[2];
  const float* Wk   = (const float*)d_in[3];
  const float* bk   = (const float*)d_in[4];
  const float* Wv   = (const float*)d_in[5];
  const float* bv   = (const float*)d_in[6];
  const float* Wo   = (const float*)d_in[7];
  const float* bo   = (const float*)d_in[8];
  const float* pos  = (const float*)d_in[9];
  const float* gWq  = (const float*)d_in[10];
  const float* gbq  = (const float*)d_in[11];
  const float* gWk  = (const float*)d_in[12];
  const float* gbk  = (const float*)d_in[13];
  const float* gWv  = (const float*)d_in[14];
  const float* gbv  = (const float*)d_in[15];
  const float* gWo  = (const float*)d_in[16];
  const float* gbo  = (const float*)d_in[17];
  const float* gng  = (const float*)d_in[18];
  const float* gnb  = (const float*)d_in[19];
  const float* W1   = (const float*)d_in[20];
  const float* b1   = (const float*)d_in[21];
  const float* W2   = (const float*)d_in[22];
  const float* b2   = (const float*)d_in[23];
  const float* ln1g = (const float*)d_in[24];
  const float* ln1b = (const float*)d_in[25];
  const float* ln2g = (const float*)d_in[26];
  const float* ln2b = (const float*)d_in[27];
  (void)in_sizes; (void)n_in; (void)out_size; (void)ws_size;

  const size_t SLOT = 800u * 192u;   // 153600 floats
  float* ws   = (float*)d_ws;
  float* Xg   = ws + 0 * SLOT;
  float* Q2   = ws + 1 * SLOT;
  float* K2   = ws + 2 * SLOT;
  float* V2   = ws + 3 * SLOT;
  float* Ctx  = ws + 4 * SLOT;
  float* Att2 = Q2;   // Q2 dead after graph_attn
  float* Xg1  = Ctx;  // Ctx dead after output projection
  float* Y1   = K2;   // K2 dead after graph_attn
  float* Y2   = V2;   // V2 dead after graph_attn

  // Stage A: chunk MHA + mean + GroupNorm + pos_emb -> Xg[800,192]
  chunk_mha_gnorm<<<dim3(800), dim3(192), 0, stream>>>(
      x, Wq, bq, Wk, bk, Wv, bv, Wo, bo, pos, gng, gnb, Xg);

  // Stage B: graph MHA (batch folded into M=800)
  dim3 gemm_grid(3, 25);   // N=192 -> 3 x 64-col panels, M=800 -> 25 x 32-row blocks
  wmma_gemm_bias<0><<<gemm_grid, 64, 0, stream>>>(Xg, gWq, gbq, Q2, 800, 192, 192);
  wmma_gemm_bias<0><<<gemm_grid, 64, 0, stream>>>(Xg, gWk, gbk, K2, 800, 192, 192);
  wmma_gemm_bias<0><<<gemm_grid, 64, 0, stream>>>(Xg, gWv, gbv, V2, 800, 192, 192);

  graph_attn<<<dim3(13, 4, 4), 32, 0, stream>>>(Q2, K2, V2, Ctx);

  wmma_gemm_bias<0><<<gemm_grid, 64, 0, stream>>>(Ctx, gWo, gbo, Att2, 800, 192, 192);
  resid_layernorm<<<dim3(800), dim3(192), 0, stream>>>(Xg, Att2, ln1g, ln1b, Xg1);

  // FFN
  wmma_gemm_bias<0><<<gemm_grid, 64, 0, stream>>>(Xg1, W1, b1, Y1, 800, 192, 192);
  wmma_gemm_bias<1><<<gemm_grid, 64, 0, stream>>>(Y1, W2, b2, Y2, 800, 192, 192);
  resid_layernorm<<<dim3(800), dim3(192), 0, stream>>>(Xg1, Y2, ln2g, ln2b, (float*)d_out);
}